// TGN1_75522704933283
// MI455X (gfx1250) — compile-verified
//
#include <hip/hip_runtime.h>
#include <hip/hip_bf16.h>

typedef __attribute__((ext_vector_type(2))) float v2f;
typedef __attribute__((ext_vector_type(8))) float v8f;
typedef __attribute__((ext_vector_type(4))) unsigned tdm4;
typedef __attribute__((ext_vector_type(8))) unsigned tdm8;

#define N_NODES 120000
#define N_EDGES 1920000
#define SEQ_LEN 30
#define NSEQ    (N_NODES / SEQ_LEN)   // 4000

// ---------------- utility kernels ----------------

__global__ void zero_f32(float* __restrict__ p, long long n) {
  long long i = (long long)blockIdx.x * blockDim.x + threadIdx.x;
  if (i < n) p[i] = 0.0f;
}

__global__ void deg_kernel(const int* __restrict__ ei, const float* __restrict__ ea,
                           float* __restrict__ deg) {
  int e = blockIdx.x * blockDim.x + threadIdx.x;
  if (e >= N_EDGES) return;
  int c = ei[N_EDGES + e];                 // col
  atomicAdd(&deg[c], ea[(long long)e * 4]); // edge_attr[:,0]
}

__global__ void dinv_kernel(const float* __restrict__ deg, float* __restrict__ dinv) {
  int i = blockIdx.x * blockDim.x + threadIdx.x;
  if (i >= N_NODES) return;
  dinv[i] = rsqrtf(deg[i] + 1.0f);         // deg >= 0, +1 self loop -> always > 0
}

// Build BigW[256][512]: rows 0..127 = Whh^T, rows 128..255 = Wih^T
__global__ void stack_weights(const float* __restrict__ Whh, const float* __restrict__ Wih,
                              float* __restrict__ BigW) {
  int tid = blockIdx.x * blockDim.x + threadIdx.x;
  if (tid >= 256 * 512) return;
  int r = tid >> 9, n = tid & 511;
  BigW[tid] = (r < 128) ? Whh[n * 128 + r] : Wih[n * 128 + (r - 128)];
}

__global__ void bias_sum(const float* __restrict__ bih, const float* __restrict__ bhh,
                         float* __restrict__ bias) {
  int i = blockIdx.x * blockDim.x + threadIdx.x;
  if (i < 512) bias[i] = bih[i] + bhh[i];
}

// ---------------- WMMA f32 GEMM: C[M,N] = A[M,K] @ B[K,N] ----------------
// Block = 256 threads (8 waves). Block computes a 32(M) x 128(N) tile:
// each wave owns one 16-col slice with TWO row-tiles (lr and lr+16), so every
// B fragment load feeds two WMMAs. A tile (contiguous 32*K floats) is moved
// into LDS by the Tensor Data Mover (wave 0 issues, waits TENSORcnt, barrier).
// K = 1<<kshift (64 or 128), N multiple of 128, M multiple of 32.
__global__ __launch_bounds__(256)
void gemm_wmma(const float* __restrict__ A, const float* __restrict__ B,
               float* __restrict__ C, int N, int kshift) {
  __shared__ float As[32 * 128];
  const int K = 1 << kshift;
  const int tid = threadIdx.x;
  const int m0 = blockIdx.x * 32;
  const int nblk = blockIdx.y * 128;

  if (tid < 32) {
    // ---- TDM: async DMA of the contiguous 32xK fp32 tile into LDS ----
    unsigned lds_off = (unsigned)(uintptr_t)(&As[0]);   // flat LDS addr low 32b = LDS byte offset
    unsigned long long ga =
        (unsigned long long)(uintptr_t)(A + (((long long)m0) << kshift));
    tdm4 g0;
    g0[0] = 1u;                                          // count=1, user descriptor
    g0[1] = lds_off;                                     // lds_addr
    g0[2] = (unsigned)(ga & 0xffffffffu);                // global_addr[31:0]
    g0[3] = (unsigned)((ga >> 32) & 0x01ffffffu) | 0x80000000u; // addr[56:32] | type=2
    tdm8 g1;
    g1[0] = 0x00020000u;                                 // wg_mask=0, data_size=2 (4B)
    g1[1] = ((unsigned)K & 0xffffu) << 16;               // tensor_dim0[15:0]
    g1[2] = (32u << 16);                                 // tensor_dim0[31:16]=0 | tensor_dim1[15:0]=32
    g1[3] = ((unsigned)K) << 16;                         // tensor_dim1 hi=0 | tile_dim0=K
    g1[4] = 32u;                                         // tile_dim1=32, tile_dim2=0
    g1[5] = (unsigned)K;                                 // tensor_dim0_stride[31:0]=K
    g1[6] = 0u;                                          // stride hi, dim1_stride lo
    g1[7] = 0u;
    asm volatile("tensor_load_to_lds %0, %1" :: "s"(g0), "s"(g1) : "memory");
    __builtin_amdgcn_s_wait_tensorcnt(0);
  }
  __syncthreads();

  const int wave = tid >> 5, lane = tid & 31;
  const int lr = lane & 15, hi = lane >> 4;
  const int n0 = nblk + wave * 16;

  v8f acc0 = {}, acc1 = {};
#pragma unroll 4
  for (int k = 0; k < K; k += 4) {
    v2f b, a0, a1;
    b.x = B[(long long)(k + 2 * hi) * N + n0 + lr];
    b.y = B[(long long)(k + 2 * hi + 1) * N + n0 + lr];
    a0.x = As[(lr << kshift) + k + 2 * hi];
    a0.y = As[(lr << kshift) + k + 2 * hi + 1];
    a1.x = As[((lr + 16) << kshift) + k + 2 * hi];
    a1.y = As[((lr + 16) << kshift) + k + 2 * hi + 1];
    acc0 = __builtin_amdgcn_wmma_f32_16x16x4_f32(false, a0, false, b, (short)0, acc0,
                                                 false, false);
    acc1 = __builtin_amdgcn_wmma_f32_16x16x4_f32(false, a1, false, b, (short)0, acc1,
                                                 false, false);
  }
#pragma unroll
  for (int v = 0; v < 8; v++) {
    int r = m0 + v + 8 * hi;
    C[(long long)r * N + n0 + lr] = acc0[v];
    C[(long long)(r + 16) * N + n0 + lr] = acc1[v];
  }
}

// ---------------- edge scatter-add (L2-resident target) ----------------
// one (edge, 4-feature chunk) per thread: E*32 threads
__global__ void edge_agg(const int* __restrict__ ei, const float* __restrict__ ea,
                         const float* __restrict__ dinv, const float* __restrict__ xw,
                         float* __restrict__ agg) {
  long long tid = (long long)blockIdx.x * blockDim.x + threadIdx.x;
  int e = (int)(tid >> 5);
  if (e >= N_EDGES) return;
  int f0 = ((int)tid & 31) * 4;
  int r = ei[e];
  int c = ei[N_EDGES + e];
  float w = ea[(long long)e * 4];
  float norm = dinv[r] * w * dinv[c];
  float4 x = *(const float4*)(xw + (long long)r * 128 + f0);
  float* dst = agg + (long long)c * 128 + f0;
  atomicAdd(dst + 0, x.x * norm);
  atomicAdd(dst + 1, x.y * norm);
  atomicAdd(dst + 2, x.z * norm);
  atomicAdd(dst + 3, x.w * norm);
}

// h = relu(agg + xw*dinv^2 + b), written in place over agg
__global__ void combine_relu(float* __restrict__ agg, const float* __restrict__ xw,
                             const float* __restrict__ dinv, const float* __restrict__ bias) {
  long long tid = (long long)blockIdx.x * blockDim.x + threadIdx.x;
  if (tid >= (long long)N_NODES * 128) return;
  int i = (int)(tid >> 7), f = (int)tid & 127;
  float d = dinv[i];
  float v = agg[tid] + xw[tid] * d * d + bias[f];
  agg[tid] = fmaxf(v, 0.0f);
}

// ---------------- fused LSTM: 16 sequences per workgroup, 30 steps in-LDS ----------------
// g[16,512] = [h(16x128) | x_t(16x128)] @ BigW[256,512] + bias; gates in LDS.
// Each wave owns 64 output columns (4 accumulators): one A-fragment load feeds 4 WMMAs.
__global__ __launch_bounds__(256)
void lstm_wmma(const float* __restrict__ x /*[120000,128] (=h2)*/,
               const float* __restrict__ BigW /*[256,512]*/,
               const float* __restrict__ bias /*[512]*/,
               float* __restrict__ h_out /*[4000,128]*/) {
  __shared__ float hbuf[16 * 128];
  __shared__ float cbuf[16 * 128];
  __shared__ float gbuf[16 * 512];
  const int tid = threadIdx.x;
  const int s0 = blockIdx.x * 16;

  for (int idx = tid; idx < 16 * 128; idx += 256) { hbuf[idx] = 0.0f; cbuf[idx] = 0.0f; }
  __syncthreads();

  const int wave = tid >> 5, lane = tid & 31;
  const int lr = lane & 15, hi = lane >> 4;
  const int nbase = wave * 64;

  for (int t = 0; t < SEQ_LEN; t++) {
    const float* xr = x + ((long long)(s0 + lr) * SEQ_LEN + t) * 128;
    v8f acc0 = {}, acc1 = {}, acc2 = {}, acc3 = {};

    // recurrent half: A = h (LDS), B = BigW rows 0..127
#pragma unroll 2
    for (int k = 0; k < 128; k += 4) {
      v2f a, b;
      a.x = hbuf[lr * 128 + k + 2 * hi];
      a.y = hbuf[lr * 128 + k + 2 * hi + 1];
      const float* b0p = BigW + (k + 2 * hi) * 512 + nbase + lr;
      const float* b1p = BigW + (k + 2 * hi + 1) * 512 + nbase + lr;
      b.x = b0p[0];  b.y = b1p[0];
      acc0 = __builtin_amdgcn_wmma_f32_16x16x4_f32(false, a, false, b, (short)0, acc0, false, false);
      b.x = b0p[16]; b.y = b1p[16];
      acc1 = __builtin_amdgcn_wmma_f32_16x16x4_f32(false, a, false, b, (short)0, acc1, false, false);
      b.x = b0p[32]; b.y = b1p[32];
      acc2 = __builtin_amdgcn_wmma_f32_16x16x4_f32(false, a, false, b, (short)0, acc2, false, false);
      b.x = b0p[48]; b.y = b1p[48];
      acc3 = __builtin_amdgcn_wmma_f32_16x16x4_f32(false, a, false, b, (short)0, acc3, false, false);
    }
    // input half: A = x_t (global, L2-resident), B = BigW rows 128..255
#pragma unroll 2
    for (int k = 0; k < 128; k += 4) {
      v2f a, b;
      a.x = xr[k + 2 * hi];
      a.y = xr[k + 2 * hi + 1];
      const float* b0p = BigW + (128 + k + 2 * hi) * 512 + nbase + lr;
      const float* b1p = BigW + (128 + k + 2 * hi + 1) * 512 + nbase + lr;
      b.x = b0p[0];  b.y = b1p[0];
      acc0 = __builtin_amdgcn_wmma_f32_16x16x4_f32(false, a, false, b, (short)0, acc0, false, false);
      b.x = b0p[16]; b.y = b1p[16];
      acc1 = __builtin_amdgcn_wmma_f32_16x16x4_f32(false, a, false, b, (short)0, acc1, false, false);
      b.x = b0p[32]; b.y = b1p[32];
      acc2 = __builtin_amdgcn_wmma_f32_16x16x4_f32(false, a, false, b, (short)0, acc2, false, false);
      b.x = b0p[48]; b.y = b1p[48];
      acc3 = __builtin_amdgcn_wmma_f32_16x16x4_f32(false, a, false, b, (short)0, acc3, false, false);
    }

#pragma unroll
    for (int v = 0; v < 8; v++) {
      int row = (v + 8 * hi) * 512 + nbase + lr;
      gbuf[row +  0] = acc0[v] + bias[nbase + lr +  0];
      gbuf[row + 16] = acc1[v] + bias[nbase + lr + 16];
      gbuf[row + 32] = acc2[v] + bias[nbase + lr + 32];
      gbuf[row + 48] = acc3[v] + bias[nbase + lr + 48];
    }
    __syncthreads();

    // ---- gate phase: i,f,g,o -> c,h (torch split order) ----
    for (int idx = tid; idx < 16 * 128; idx += 256) {
      int s = idx >> 7, j = idx & 127;
      float ig = gbuf[s * 512 + j];
      float fg = gbuf[s * 512 + 128 + j];
      float gg = gbuf[s * 512 + 256 + j];
      float og = gbuf[s * 512 + 384 + j];
      float si = 1.0f / (1.0f + __expf(-ig));
      float sf = 1.0f / (1.0f + __expf(-fg));
      float so = 1.0f / (1.0f + __expf(-og));
      float c = sf * cbuf[idx] + si * tanhf(gg);
      cbuf[idx] = c;
      hbuf[idx] = so * tanhf(c);
    }
    __syncthreads();
  }

  for (int idx = tid; idx < 16 * 128; idx += 256) {
    int s = idx >> 7, j = idx & 127;
    h_out[(long long)(s0 + s) * 128 + j] = hbuf[idx];
  }
}

// ---------------- FC1+ReLU+FC2+softmax, one row per 64-thread block ----------------
__global__ void fc_softmax(const float* __restrict__ h_all, const float* __restrict__ fcW1,
                           const float* __restrict__ fcb1, const float* __restrict__ fcW2,
                           const float* __restrict__ fcb2, float* __restrict__ out) {
  __shared__ float fc1[64];
  __shared__ float lg[3];
  int r = blockIdx.x;
  int j = threadIdx.x;
  const float* hr = h_all + (long long)r * 128;
  float s = fcb1[j];
  for (int k = 0; k < 128; k++) s += hr[k] * fcW1[j * 128 + k];
  fc1[j] = fmaxf(s, 0.0f);
  __syncthreads();
  if (j < 3) {
    float s2 = fcb2[j];
    for (int k = 0; k < 64; k++) s2 += fc1[k] * fcW2[j * 64 + k];
    lg[j] = s2;
  }
  __syncthreads();
  if (j == 0) {
    float m = fmaxf(lg[0], fmaxf(lg[1], lg[2]));
    float e0 = __expf(lg[0] - m), e1 = __expf(lg[1] - m), e2 = __expf(lg[2] - m);
    float inv = 1.0f / (e0 + e1 + e2);
    out[(long long)r * 3 + 0] = e0 * inv;
    out[(long long)r * 3 + 1] = e1 * inv;
    out[(long long)r * 3 + 2] = e2 * inv;
  }
}

// ---------------- driver ----------------

extern "C" void kernel_launch(void* const* d_in, const int* in_sizes, int n_in,
                              void* d_out, int out_size, void* d_ws, size_t ws_size,
                              hipStream_t stream) {
  const float* x    = (const float*)d_in[0];
  const int*   ei   = (const int*)  d_in[1];
  const float* ea   = (const float*)d_in[2];
  const float* W1   = (const float*)d_in[3];
  const float* b1   = (const float*)d_in[4];
  const float* W2   = (const float*)d_in[5];
  const float* b2   = (const float*)d_in[6];
  const float* Wih  = (const float*)d_in[7];
  const float* Whh  = (const float*)d_in[8];
  const float* bih  = (const float*)d_in[9];
  const float* bhh  = (const float*)d_in[10];
  const float* fcW1 = (const float*)d_in[11];
  const float* fcb1 = (const float*)d_in[12];
  const float* fcW2 = (const float*)d_in[13];
  const float* fcb2 = (const float*)d_in[14];
  float* out = (float*)d_out;

  const long long NF = (long long)N_NODES * 128;
  float* ws   = (float*)d_ws;
  float* bufA = ws;                       // agg1 -> h1 -> agg2 -> h2
  float* bufB = bufA + NF;                // xw1 -> xw2
  float* deg  = bufB + NF;
  float* dinv = deg + N_NODES;
  float* BigW = dinv + N_NODES;           // [256,512]
  float* bias = BigW + 256 * 512;         // [512]
  float* hall = bias + 512;               // [4000,128]

  dim3 b256(256);
  int nodeBlocks = (N_NODES + 255) / 256;
  int nfBlocks   = (int)((NF + 255) / 256);
  int edgeBlocks = (N_EDGES + 255) / 256;
  int aggBlocks  = (int)(((long long)N_EDGES * 32 + 255) / 256);

  // degrees + sym-norm factors
  zero_f32<<<nodeBlocks, b256, 0, stream>>>(deg, N_NODES);
  deg_kernel<<<edgeBlocks, b256, 0, stream>>>(ei, ea, deg);
  dinv_kernel<<<nodeBlocks, b256, 0, stream>>>(deg, dinv);

  // LSTM weight prep
  stack_weights<<<(256 * 512 + 255) / 256, b256, 0, stream>>>(Whh, Wih, BigW);
  bias_sum<<<2, b256, 0, stream>>>(bih, bhh, bias);

  // ---- GCN layer 1 ----
  gemm_wmma<<<dim3(N_NODES / 32, 1), b256, 0, stream>>>(x, W1, bufB, 128, 6);   // xw1 = x@W1
  zero_f32<<<nfBlocks, b256, 0, stream>>>(bufA, NF);
  edge_agg<<<aggBlocks, b256, 0, stream>>>(ei, ea, dinv, bufB, bufA);
  combine_relu<<<nfBlocks, b256, 0, stream>>>(bufA, bufB, dinv, b1);            // h1 in bufA

  // ---- GCN layer 2 ----
  gemm_wmma<<<dim3(N_NODES / 32, 1), b256, 0, stream>>>(bufA, W2, bufB, 128, 7); // xw2 = h1@W2
  zero_f32<<<nfBlocks, b256, 0, stream>>>(bufA, NF);
  edge_agg<<<aggBlocks, b256, 0, stream>>>(ei, ea, dinv, bufB, bufA);
  combine_relu<<<nfBlocks, b256, 0, stream>>>(bufA, bufB, dinv, b2);            // h2 in bufA

  // ---- LSTM (fused, 30 steps in-LDS) ----
  lstm_wmma<<<NSEQ / 16, b256, 0, stream>>>(bufA, BigW, bias, hall);

  // ---- FC head + softmax ----
  fc_softmax<<<NSEQ, 64, 0, stream>>>(hall, fcW1, fcb1, fcW2, fcb2, out);
}